// BaseBoxPostProcessor_66228395704893
// MI455X (gfx1250) — compile-verified
//
#include <hip/hip_runtime.h>
#include <hip/hip_bf16.h>

// ---------------------------------------------------------------------------
// BaseBoxPostProcessor for MI455X (gfx1250).
//   B=2 images, N=1024 proposals, C=61 classes, K=100 detections/img.
// CDNA5 paths used:
//   * V_WMMA_F32_16X16X4_F32 with an all-ones B matrix computes the softmax
//     denominators (16 fp32 row-sums per instruction) in K1.
//   * Tensor Data Mover (tensor_load_to_lds) performs the strided
//     class-column gathers into LDS in K2, synced via s_wait_tensorcnt.
// Workspace: ~4.96 MB float32 scratch.
// ---------------------------------------------------------------------------

#define BB   2
#define NN   1024
#define CC   61
#define CF   60          // foreground classes
#define KK   100
#define TPB  256

#define W_IMG 1024.0f
#define H_IMG 1024.0f
#define SCORE_T 0.05f
#define NMS_T   0.5f
#define SCALE_CLAMP 4.135166556742356f   // log(1000/16)

typedef unsigned int u32x4 __attribute__((ext_vector_type(4)));
typedef int          i32x4 __attribute__((ext_vector_type(4)));
typedef int          i32x8 __attribute__((ext_vector_type(8)));
typedef float        v2f   __attribute__((ext_vector_type(2)));
typedef float        v8f   __attribute__((ext_vector_type(8)));

// Build a 2D-tile Tensor DMA descriptor (D#) and issue TENSOR_LOAD_TO_LDS.
//   element = 4 bytes (data_size code 2)
//   tile   = tile_d0 x tile_d1 elements, row stride = stride0 elements
__device__ __forceinline__ void tdm_load_2d(const void* gptr, void* lptr,
                                            unsigned tensor_d0, unsigned tensor_d1,
                                            unsigned tile_d0, unsigned tile_d1,
                                            unsigned long long stride0) {
  unsigned long long ga = (unsigned long long)(uintptr_t)gptr;
  unsigned la = (unsigned)(uintptr_t)lptr;   // low 32 bits of flat ptr = LDS addr
  u32x4 g0;
  g0.x = 1u;                                            // count=1, user descriptor
  g0.y = la;                                            // lds_addr [63:32]
  g0.z = (unsigned)(ga & 0xffffffffull);                // global_addr[31:0]
  g0.w = (unsigned)((ga >> 32) & 0x1ffffffull)          // global_addr[56:32]
       | 0x80000000u;                                   // type = 2 ("image")
  i32x8 g1;
  g1[0] = (int)(2u << 16);                              // data_size = 4 bytes
  g1[1] = (int)((tensor_d0 & 0xffffu) << 16);           // tensor_dim0[15:0] @ bit48
  g1[2] = (int)(((tensor_d0 >> 16) & 0xffffu) |
                ((tensor_d1 & 0xffffu) << 16));
  g1[3] = (int)(((tensor_d1 >> 16) & 0xffffu) |
                ((tile_d0 & 0xffffu) << 16));
  g1[4] = (int)(tile_d1 & 0xffffu);                     // tile_dim1 (tile_dim2=0)
  g1[5] = (int)(unsigned)(stride0 & 0xffffffffull);     // tensor_dim0_stride[31:0]
  g1[6] = (int)(unsigned)((stride0 >> 32) & 0xffffull); // stride[47:32], stride1=0
  g1[7] = 0;
  i32x4 z4 = {0, 0, 0, 0};
#if __clang_major__ >= 23
  i32x8 z8 = {0, 0, 0, 0, 0, 0, 0, 0};
  __builtin_amdgcn_tensor_load_to_lds(g0, g1, z4, z4, z8, 0);
#else
  __builtin_amdgcn_tensor_load_to_lds(g0, g1, z4, z4, 0);
#endif
}

// ---------------------------------------------------------------------------
// K1: softmax + per-class box decode/clip.
// Each wave owns 16 proposals. Softmax denominators computed on the matrix
// pipe: D = A(16x4 exp-chunk) x B(4x16 ones) + C accumulates fp32 row sums,
// replicated across D's columns (V_WMMA_F32_16X16X4_F32).
//   A layout (ISA 7.12.2): lanes 0-15 hold K=0,1; lanes 16-31 hold K=2,3.
//   D layout: VGPR v / lanes 0-15 -> row v ; lanes 16-31 -> row v+8.
// Then the two lane-halves split the 61 classes for probs + decode.
//   probs  : [B][N][C]       boxes : [B][N][C][4]
// ---------------------------------------------------------------------------
__global__ __launch_bounds__(TPB)
void k_softmax_decode(const float* __restrict__ logits,
                      const float* __restrict__ deltas,
                      const float* __restrict__ props,
                      float* __restrict__ probs,
                      float* __restrict__ boxes) {
  __shared__ float wsum[TPB / 32][16];

  const int tid  = threadIdx.x;
  const int wid  = tid >> 5;          // wave in block
  const int lane = tid & 31;
  const int r    = lane & 15;         // row within the wave's 16-proposal tile
  const int half = lane >> 4;         // 0: K=0,1 / cols 0..30 ; 1: K=2,3 / 31..60
  const int gw   = blockIdx.x * (TPB / 32) + wid;   // global wave id
  const int p    = gw * 16 + r;       // proposal handled by this lane (dup/half)

  const float* lrow = logits + (size_t)p * CC;

  // --- row sums of exp(logits) via WMMA (no max-subtract: |logits| ~ 8 max,
  //     exp() is far from fp32 overflow; mathematically identical to softmax)
  v8f acc = {};
  v2f bones; bones.x = 1.0f; bones.y = 1.0f;
  for (int t = 0; t < 15; ++t) {                       // cols 0..59
    int col = 4 * t + 2 * half;
    v2f a;
    a.x = __expf(lrow[col]);
    a.y = __expf(lrow[col + 1]);
    acc = __builtin_amdgcn_wmma_f32_16x16x4_f32(
        false, a, false, bones, (short)0, acc, false, false);
  }
  {                                                    // remainder col 60
    float e60 = __expf(lrow[60]);                      // always in-bounds
    v2f a;
    a.x = (half == 0) ? e60 : 0.0f;
    a.y = 0.0f;
    acc = __builtin_amdgcn_wmma_f32_16x16x4_f32(
        false, a, false, bones, (short)0, acc, false, false);
  }
  // publish the 16 row sums (column 0 of D): lane 0 -> rows 0..7, lane 16 -> 8..15
  if (lane == 0) {
    for (int v = 0; v < 8; ++v) wsum[wid][v] = acc[v];
  } else if (lane == 16) {
    for (int v = 0; v < 8; ++v) wsum[wid][8 + v] = acc[v];
  }
  __syncthreads();
  const float inv = 1.0f / wsum[wid][r];

  // --- per-lane probs + decode; the two halves split the class range
  const float* pr = props + (size_t)p * 4;
  float x1 = pr[0], y1 = pr[1], x2 = pr[2], y2 = pr[3];
  float w = x2 - x1, h = y2 - y1;
  float cx = x1 + 0.5f * w, cy = y1 + 0.5f * h;

  const float* drow = deltas + (size_t)p * CC * 4;
  const int c0 = (half == 0) ? 0 : 31;
  const int c1 = (half == 0) ? 31 : CC;
  for (int c = c0; c < c1; ++c) {
    probs[(size_t)p * CC + c] = __expf(lrow[c]) * inv;
    float dx = drow[4 * c + 0] * 0.1f;                 // / WX
    float dy = drow[4 * c + 1] * 0.1f;                 // / WY
    float dw = fminf(drow[4 * c + 2] * 0.2f, SCALE_CLAMP);
    float dh = fminf(drow[4 * c + 3] * 0.2f, SCALE_CLAMP);
    float pcx = dx * w + cx, pcy = dy * h + cy;
    float pw = __expf(dw) * w, ph = __expf(dh) * h;
    float bx1 = fminf(fmaxf(pcx - 0.5f * pw, 0.0f), W_IMG);
    float by1 = fminf(fmaxf(pcy - 0.5f * ph, 0.0f), H_IMG);
    float bx2 = fminf(fmaxf(pcx + 0.5f * pw, 0.0f), W_IMG);
    float by2 = fminf(fmaxf(pcy + 0.5f * ph, 0.0f), H_IMG);
    size_t o = ((size_t)p * CC + c) * 4;
    boxes[o + 0] = bx1; boxes[o + 1] = by1;
    boxes[o + 2] = bx2; boxes[o + 3] = by2;
  }
}

// ---------------------------------------------------------------------------
// K2: one block per (image, fg class): TDM gather -> sort -> greedy NMS.
// Results stored in score-sorted order (matches reference flattening):
//   s_k : [B][CF][N]    b_k : [B][CF][N][4]
// ---------------------------------------------------------------------------
__global__ __launch_bounds__(TPB)
void k_nms(const float* __restrict__ probs, const float* __restrict__ boxes,
           float* __restrict__ s_k, float* __restrict__ b_k) {
  int blk = blockIdx.x;
  int b = blk / CF;
  int j = blk % CF;
  int cls = j + 1;
  int tid = threadIdx.x;

  __shared__ float scr[NN];            // raw class scores (TDM dest)
  __shared__ float bxRaw[NN * 4];      // raw class boxes  (TDM dest)
  __shared__ float s[NN];              // sort keys  (score or -1)
  __shared__ int   sidx[NN];           // original index
  __shared__ float bx[NN * 4];         // boxes in sorted order
  __shared__ unsigned char sup[NN];
  __shared__ unsigned char keepf[NN];
  __shared__ int nv_sh;

  // --- Tensor Data Mover: gather strided class column + box tile into LDS.
  int wid = __builtin_amdgcn_readfirstlane((int)(threadIdx.x >> 5));
  if (wid == 0) {
    // scores: tensor [N rows][C cols] f32, tile 1x1024, row stride C
    tdm_load_2d(probs + (size_t)b * NN * CC + cls, scr,
                CC, NN, 1, NN, (unsigned long long)CC);
    // boxes: tensor [N rows][C*4 cols] f32, tile 4x1024, row stride C*4
    tdm_load_2d(boxes + ((size_t)b * NN * CC + (size_t)cls) * 4, bxRaw,
                CC * 4, NN, 4, NN, (unsigned long long)(CC * 4));
    __builtin_amdgcn_s_wait_tensorcnt(0);
  }
  __syncthreads();

  // --- init keys / flags
  if (tid == 0) nv_sh = 0;
  for (int n = tid; n < NN; n += TPB) {
    float p = scr[n];
    s[n] = (p > SCORE_T) ? p : -1.0f;   // matches key_s = where(valid, s, -1)
    sidx[n] = n;
    sup[n] = 0;
    keepf[n] = 0;
  }
  __syncthreads();

  // --- bitonic sort: score descending, ties by original index ascending
  for (int k = 2; k <= NN; k <<= 1) {
    for (int jj = k >> 1; jj > 0; jj >>= 1) {
      for (int i = tid; i < NN; i += TPB) {
        int ixj = i ^ jj;
        if (ixj > i) {
          float si = s[i], sj = s[ixj];
          int ii = sidx[i], ij = sidx[ixj];
          bool iGreater = (si > sj) || (si == sj && ii < ij);
          bool desc = ((i & k) == 0);
          if (desc ? !iGreater : iGreater) {
            s[i] = sj;   s[ixj] = si;
            sidx[i] = ij; sidx[ixj] = ii;
          }
        }
      }
      __syncthreads();
    }
  }

  // --- count valid (sorted to front) + gather boxes into sorted order
  int cnt = 0;
  for (int n = tid; n < NN; n += TPB) {
    cnt += (s[n] > SCORE_T) ? 1 : 0;
    int o = sidx[n] * 4;
    bx[4 * n + 0] = bxRaw[o + 0];
    bx[4 * n + 1] = bxRaw[o + 1];
    bx[4 * n + 2] = bxRaw[o + 2];
    bx[4 * n + 3] = bxRaw[o + 3];
  }
  if (cnt) atomicAdd(&nv_sh, cnt);
  __syncthreads();
  int nv = nv_sh;

  // --- greedy NMS (one barrier per candidate; self-suppress skipped so the
  //     uniform read of sup[i] never races with this iteration's writes)
  for (int i = 0; i < nv; ++i) {
    __syncthreads();
    if (sup[i]) continue;                      // uniform across block
    if (tid == 0) keepf[i] = 1;
    float ax1 = bx[4 * i + 0], ay1 = bx[4 * i + 1];
    float ax2 = bx[4 * i + 2], ay2 = bx[4 * i + 3];
    float aA = (ax2 - ax1) * (ay2 - ay1);
    for (int jn = tid; jn < nv; jn += TPB) {
      if (jn == i) continue;
      float q1 = bx[4 * jn + 0], q2 = bx[4 * jn + 1];
      float q3 = bx[4 * jn + 2], q4 = bx[4 * jn + 3];
      float bA = (q3 - q1) * (q4 - q2);
      float iw = fminf(ax2, q3) - fmaxf(ax1, q1);
      float ih = fminf(ay2, q4) - fmaxf(ay1, q2);
      iw = fmaxf(iw, 0.0f); ih = fmaxf(ih, 0.0f);
      float inter = iw * ih;
      float iou = inter / (aA + bA - inter + 1e-9f);
      if (iou > NMS_T) sup[jn] = 1;
    }
  }
  __syncthreads();

  // --- emit per-class results in sorted order
  size_t base = ((size_t)b * CF + (size_t)j) * NN;
  for (int n = tid; n < NN; n += TPB) {
    bool kp = keepf[n] != 0;
    s_k[base + n] = kp ? s[n] : 0.0f;
    size_t o = (base + n) * 4;
    b_k[o + 0] = kp ? bx[4 * n + 0] : 0.0f;
    b_k[o + 1] = kp ? bx[4 * n + 1] : 0.0f;
    b_k[o + 2] = kp ? bx[4 * n + 2] : 0.0f;
    b_k[o + 3] = kp ? bx[4 * n + 3] : 0.0f;
  }
}

// ---------------------------------------------------------------------------
// K3: per-image top-K with lax.top_k tie semantics (smallest flat index wins
// ties, including the zero-score filler entries). Destructive marking of s_k
// (regenerated by K2 every call -> deterministic).
// ---------------------------------------------------------------------------
__global__ __launch_bounds__(TPB)
void k_topk(float* __restrict__ s_k, const float* __restrict__ b_k,
            float* __restrict__ dets, int* __restrict__ labels) {
  int b = blockIdx.x;
  int tid = threadIdx.x;
  const int M = CF * NN;
  float* sArr = s_k + (size_t)b * M;
  const float* bArr = b_k + (size_t)b * M * 4;
  __shared__ float rs[TPB];
  __shared__ int   ri[TPB];

  for (int kk = 0; kk < KK; ++kk) {
    float best = -3.0e38f;
    int   bi = M;
    for (int m = tid; m < M; m += TPB) {
      float v = sArr[m];
      if (v > best) { best = v; bi = m; }   // strict > keeps smallest m on ties
    }
    rs[tid] = best; ri[tid] = bi;
    __syncthreads();
    if (tid == 0) {
      float bv = rs[0]; int bix = ri[0];
      for (int t = 1; t < TPB; ++t) {
        if (rs[t] > bv || (rs[t] == bv && ri[t] < bix)) { bv = rs[t]; bix = ri[t]; }
      }
      float* dr = dets + ((size_t)b * KK + kk) * 5;
      dr[0] = bArr[4 * bix + 0];
      dr[1] = bArr[4 * bix + 1];
      dr[2] = bArr[4 * bix + 2];
      dr[3] = bArr[4 * bix + 3];
      dr[4] = bv;
      labels[b * KK + kk] = bix / NN + 1;
      sArr[bix] = -3.0e38f;                 // exclude from later passes
    }
    __syncthreads();                         // fences the global mark too
  }
}

// ---------------------------------------------------------------------------
extern "C" void kernel_launch(void* const* d_in, const int* in_sizes, int n_in,
                              void* d_out, int out_size, void* d_ws, size_t ws_size,
                              hipStream_t stream) {
  (void)in_sizes; (void)n_in; (void)out_size; (void)ws_size;
  const float* logits = (const float*)d_in[0];   // [B*N, C]
  const float* deltas = (const float*)d_in[1];   // [B*N, C*4]
  const float* props  = (const float*)d_in[2];   // [B, N, 4]

  float* ws    = (float*)d_ws;
  float* probs = ws;                         // B*N*C      = 124928
  float* boxes = ws + 124928;                // B*N*C*4    = 499712
  float* s_k   = ws + 624640;                // B*CF*N     = 122880
  float* b_k   = ws + 747520;                // B*CF*N*4   = 491520
                                             // total ~4.96 MB

  float* dets   = (float*)d_out;             // [B, K, 5]
  int*   labels = (int*)d_out + BB * KK * 5; // [B, K] int32 bits

  // 16 proposals per wave, 8 waves per block -> 16 blocks for B*N = 2048
  k_softmax_decode<<<(BB * NN) / (16 * (TPB / 32)), TPB, 0, stream>>>(
      logits, deltas, props, probs, boxes);
  k_nms<<<BB * CF, TPB, 0, stream>>>(probs, boxes, s_k, b_k);
  k_topk<<<BB, TPB, 0, stream>>>(s_k, b_k, dets, labels);
}